// FeatureGraphAttention_22728966930820
// MI455X (gfx1250) — compile-verified
//
#include <hip/hip_runtime.h>
#include <hip/hip_bf16.h>
#include <math.h>

typedef __attribute__((ext_vector_type(2))) float v2f;
typedef __attribute__((ext_vector_type(8))) float v8f;

#define GN 512
#define GB 256

// ---------------------------------------------------------------------------
// Setup: collapse the rank-1 weight structure into per-head score coefficients
// (scale folded in), output vectors u0/u1/c0, and closed-form LN statistics.
// ws layout (floats):
//  [0..1] cA(h)  [2..3] cB(h)  [4..5] cC(h)  [6..7] cD(h)
//  [8..10] Mu0, Mu1, Mc   (mean coefficients)
//  [11..16] e00,e11,ecc,e01,e0c,e1c   (E[y^2] coefficients)
//  [32..95] u0, [96..159] u1, [160..223] c0
//  [256..511] fmean[b]
// ---------------------------------------------------------------------------
__global__ void fga_setup_kernel(const float* __restrict__ We, const float* __restrict__ be,
                                 const float* __restrict__ Wq, const float* __restrict__ bq,
                                 const float* __restrict__ Wk, const float* __restrict__ bk,
                                 const float* __restrict__ Wv, const float* __restrict__ bv,
                                 const float* __restrict__ Wo, const float* __restrict__ bo,
                                 const float* __restrict__ temperature,
                                 float* __restrict__ ws) {
  __shared__ float wq[64], cq[64], wk[64], ck[64], wv[64], cv[64];
  __shared__ float u0s[64], u1s[64], c0s[64];
  const int o = threadIdx.x;  // 0..63
  float sq = 0.f, sbq = 0.f, sk = 0.f, sbk = 0.f, sv = 0.f, sbv = 0.f;
  for (int f = 0; f < 64; ++f) {
    const float we = We[f], bb = be[f];
    sq += Wq[o * 64 + f] * we;  sbq += Wq[o * 64 + f] * bb;
    sk += Wk[o * 64 + f] * we;  sbk += Wk[o * 64 + f] * bb;
    sv += Wv[o * 64 + f] * we;  sbv += Wv[o * 64 + f] * bb;
  }
  wq[o] = sq; cq[o] = sbq + bq[o];
  wk[o] = sk; ck[o] = sbk + bk[o];
  wv[o] = sv; cv[o] = sbv + bv[o];
  __syncthreads();

  float u0 = 0.f, u1 = 0.f, c0 = 0.f;
  for (int j = 0; j < 32; ++j) {
    u0 += Wo[o * 64 + j] * wv[j];
    u1 += Wo[o * 64 + 32 + j] * wv[32 + j];
  }
  for (int j = 0; j < 64; ++j) c0 += Wo[o * 64 + j] * cv[j];
  c0 += bo[o];
  u0s[o] = u0; u1s[o] = u1; c0s[o] = c0;
  ws[32 + o] = u0; ws[96 + o] = u1; ws[160 + o] = c0;

  if (o < 2) {
    const int base = o * 32;
    float a = 0.f, b2 = 0.f, c2 = 0.f, d2 = 0.f;
    for (int j = 0; j < 32; ++j) {
      a  += wq[base + j] * wk[base + j];
      b2 += wq[base + j] * ck[base + j];
      c2 += cq[base + j] * wk[base + j];
      d2 += cq[base + j] * ck[base + j];
    }
    const float scale = temperature[0] * 0.17677669529663687f;  // 1/sqrt(32)
    ws[0 + o] = scale * a;  ws[2 + o] = scale * b2;
    ws[4 + o] = scale * c2; ws[6 + o] = scale * d2;
  }
  __syncthreads();

  if (o == 0) {
    float S0 = 0.f, S1 = 0.f, Sc = 0.f;
    float Q00 = 0.f, Q11 = 0.f, Qcc = 0.f, Q01 = 0.f, Q0c = 0.f, Q1c = 0.f;
    for (int d = 0; d < 64; ++d) {
      const float a = u0s[d], b = u1s[d], c = c0s[d];
      S0 += a; S1 += b; Sc += c;
      Q00 += a * a; Q11 += b * b; Qcc += c * c;
      Q01 += a * b; Q0c += a * c; Q1c += b * c;
    }
    const float inv = 1.0f / 64.0f;
    ws[8]  = S0 * inv; ws[9]  = S1 * inv; ws[10] = Sc * inv;
    ws[11] = Q00 * inv; ws[12] = Q11 * inv; ws[13] = Qcc * inv;
    ws[14] = 2.f * Q01 * inv; ws[15] = 2.f * Q0c * inv; ws[16] = 2.f * Q1c * inv;
  }
}

// fmean[b] = mean_m features[b, m]  (uniform-softmax fallback for fully masked rows)
__global__ void fga_fmean_kernel(const float* __restrict__ feats, float* __restrict__ ws) {
  const int b = blockIdx.x;
  const int t = threadIdx.x;  // 64 threads
  float s = 0.f;
  for (int m = t; m < GN; m += 64) s += feats[b * GN + m];
  __shared__ float red[64];
  red[t] = s;
  __syncthreads();
  if (t < 32) red[t] += red[t + 32];
  __syncthreads();
  if (t == 0) {
    float tot = 0.f;
    for (int i = 0; i < 32; ++i) tot += red[i];
    ws[256 + b] = tot * (1.0f / (float)GN);
  }
}

// ---------------------------------------------------------------------------
// Main kernel: one wave handles a 16-row tile of one batch, both heads.
// Score tile via V_WMMA_F32_16X16X4_F32 (rank-2 factorization, exact),
// masked exp accumulation in the C/D accumulator layout, 16-lane row
// reduction, closed-form LayerNorm.
// ---------------------------------------------------------------------------
__global__ __launch_bounds__(256) void fga_attn_kernel(
    const float* __restrict__ feats, const int* __restrict__ adj,
    const float* __restrict__ gamma, const float* __restrict__ beta,
    const float* __restrict__ ws, float* __restrict__ out) {
  __shared__ float sf[GN];
  __shared__ float su0[64], su1[64], sc0[64], sg[64], sb[64];
  __shared__ float scoef[17];

  const int b = blockIdx.y;
  const int t = threadIdx.x;

  for (int i = t; i < GN; i += 256) sf[i] = feats[b * GN + i];
  if (t < 64) {
    su0[t] = ws[32 + t]; su1[t] = ws[96 + t]; sc0[t] = ws[160 + t];
    sg[t] = gamma[t]; sb[t] = beta[t];
  }
  if (t < 17) scoef[t] = ws[t];
  __syncthreads();

  const int wave = t >> 5;
  const int lane = t & 31;
  const int n0 = (blockIdx.x * 8 + wave) * 16;
  const int hi = lane >> 4;
  const int lo = lane & 15;

  const float cA0 = scoef[0], cA1 = scoef[1], cB0 = scoef[2], cB1 = scoef[3];
  const float cC0 = scoef[4], cC1 = scoef[5], cD0 = scoef[6], cD1 = scoef[7];

  // A fragment (16x4): col0 = f_n, col1 = 1, cols 2..3 = 0 (upper lanes hold K=2,3)
  v2f afrag;
  afrag.x = hi ? 0.0f : sf[n0 + lo];
  afrag.y = hi ? 0.0f : 1.0f;

  v8f accP0 = {}, accPF0 = {}, accP1 = {}, accPF1 = {};

  const int* abase = adj + ((size_t)b * GN + (size_t)(n0 + 8 * hi)) * GN + lo;
  const float LOG2E = 1.4426950408889634f;

  for (int m0 = 0; m0 < GN; m0 += 16) {
    const float fm = sf[m0 + lo];
    // B fragment (4x16): row0 = scale*(a*f_m + b), row1 = scale*(c*f_m + d)
    v2f b0, b1;
    b0.x = hi ? 0.0f : fmaf(cA0, fm, cB0);
    b0.y = hi ? 0.0f : fmaf(cC0, fm, cD0);
    b1.x = hi ? 0.0f : fmaf(cA1, fm, cB1);
    b1.y = hi ? 0.0f : fmaf(cC1, fm, cD1);

    v8f z = {};
    v8f s0t = __builtin_amdgcn_wmma_f32_16x16x4_f32(false, afrag, false, b0,
                                                    (short)0, z, false, false);
    v8f s1t = __builtin_amdgcn_wmma_f32_16x16x4_f32(false, afrag, false, b1,
                                                    (short)0, z, false, false);

    const int* ap = abase + m0;
#pragma unroll
    for (int v = 0; v < 8; ++v) {
      const float adjf = (float)ap[v * GN];  // 0 or 1
      const float x0 = fminf(fmaxf(s0t[v], -10.f), 10.f);
      const float x1 = fminf(fmaxf(s1t[v], -10.f), 10.f);
      const float p0 = __builtin_amdgcn_exp2f(x0 * LOG2E) * adjf;
      const float p1 = __builtin_amdgcn_exp2f(x1 * LOG2E) * adjf;
      accP0[v] += p0;  accPF0[v] += p0 * fm;
      accP1[v] += p1;  accPF1[v] += p1 * fm;
    }
  }

  // Row sums: reduce across the 16 lanes of each half-wave.
#pragma unroll
  for (int v = 0; v < 8; ++v) {
    float p0 = accP0[v], pf0 = accPF0[v], p1 = accP1[v], pf1 = accPF1[v];
#pragma unroll
    for (int m = 1; m < 16; m <<= 1) {
      p0  += __shfl_xor(p0, m, 16);
      pf0 += __shfl_xor(pf0, m, 16);
      p1  += __shfl_xor(p1, m, 16);
      pf1 += __shfl_xor(pf1, m, 16);
    }
    accP0[v] = p0; accPF0[v] = pf0; accP1[v] = p1; accPF1[v] = pf1;
  }

  const float fmean = ws[256 + b];
  const float Mu0 = scoef[8], Mu1 = scoef[9], Mc = scoef[10];
  const float e00 = scoef[11], e11 = scoef[12], ecc = scoef[13];
  const float e01 = scoef[14], e0c = scoef[15], e1c = scoef[16];

#pragma unroll
  for (int v = 0; v < 8; ++v) {
    const int row = n0 + v + 8 * hi;
    const float s0 = accP0[v] > 0.f ? accPF0[v] / accP0[v] : fmean;
    const float s1 = accP1[v] > 0.f ? accPF1[v] / accP1[v] : fmean;
    const float mu = fmaf(s0, Mu0, fmaf(s1, Mu1, Mc));
    const float ey2 = s0 * s0 * e00 + s1 * s1 * e11 + ecc +
                      s0 * s1 * e01 + s0 * e0c + s1 * e1c;
    const float var = ey2 - mu * mu;
    const float rs = rsqrtf(var + 1e-5f);
    float* op = out + ((size_t)b * GN + row) * 64;
#pragma unroll
    for (int j = 0; j < 4; ++j) {
      const int d = lo + 16 * j;
      const float y = fmaf(s0, su0[d], fmaf(s1, su1[d], sc0[d]));
      op[d] = fmaf((y - mu) * rs, sg[d], sb[d]);
    }
  }
}

extern "C" void kernel_launch(void* const* d_in, const int* in_sizes, int n_in,
                              void* d_out, int out_size, void* d_ws, size_t ws_size,
                              hipStream_t stream) {
  const float* features    = (const float*)d_in[0];
  const int*   adjacency   = (const int*)d_in[1];
  const float* We          = (const float*)d_in[2];
  const float* be          = (const float*)d_in[3];
  const float* Wq          = (const float*)d_in[4];
  const float* bq          = (const float*)d_in[5];
  const float* Wk          = (const float*)d_in[6];
  const float* bk          = (const float*)d_in[7];
  const float* Wv          = (const float*)d_in[8];
  const float* bv          = (const float*)d_in[9];
  const float* Wo          = (const float*)d_in[10];
  const float* bo          = (const float*)d_in[11];
  const float* gamma       = (const float*)d_in[12];
  const float* beta        = (const float*)d_in[13];
  const float* temperature = (const float*)d_in[14];
  float* ws  = (float*)d_ws;
  float* out = (float*)d_out;

  hipLaunchKernelGGL(fga_setup_kernel, dim3(1), dim3(64), 0, stream,
                     We, be, Wq, bq, Wk, bk, Wv, bv, Wo, bo, temperature, ws);
  hipLaunchKernelGGL(fga_fmean_kernel, dim3(GB), dim3(64), 0, stream, features, ws);
  hipLaunchKernelGGL(fga_attn_kernel, dim3(GN / (16 * 8), GB), dim3(256), 0, stream,
                     features, adjacency, gamma, beta, ws, out);
}